// scatter2dc_67594195304880
// MI455X (gfx1250) — compile-verified
//
#include <hip/hip_runtime.h>
#include <hip/hip_bf16.h>

// Problem constants from the reference: input [B=4, C=64, H=512, W=512] f32,
// gathered [N, 64, 16, 16] f32, indices [N, 3] int64, stride = kernel = 16.
#define CC 64
#define HH 512
#define WW 512
#define KH 16
#define KW 16
#define PATCH_F (CC * KH * KW)      // 16384 floats = 64 KB per patch
#define PATCH_F4 (PATCH_F / 4)      // 4096 float4
#define TPB 256

typedef int v4i __attribute__((ext_vector_type(4)));
typedef __attribute__((address_space(1))) v4i gv4i;   // global int4
typedef __attribute__((address_space(3))) v4i lv4i;   // LDS int4

// ---- CDNA5 async global->LDS copy (ASYNCcnt path), with safe fallback ----
__device__ __forceinline__ void async_g2l_b128(const float* g, float* l) {
#if __has_builtin(__builtin_amdgcn_global_load_async_to_lds_b128)
    __builtin_amdgcn_global_load_async_to_lds_b128(
        (gv4i*)const_cast<float*>(g), (lv4i*)l, /*offset=*/0, /*cpol=*/0);
#else
    *(float4*)l = *(const float4*)g;   // fallback: VGPR round trip
#endif
}

__device__ __forceinline__ void wait_async_done() {
#if __has_builtin(__builtin_amdgcn_global_load_async_to_lds_b128)
#if __has_builtin(__builtin_amdgcn_s_wait_asynccnt)
    __builtin_amdgcn_s_wait_asynccnt(0);
#else
    asm volatile("s_wait_asynccnt 0" ::: "memory");
#endif
#endif
}

// ---- Kernel 1: out = input (bandwidth-bound float4 copy) ----
__global__ __launch_bounds__(TPB) void scatter2dc_copy_kernel(
    const float4* __restrict__ in, float4* __restrict__ out, int n4) {
    int stride = (int)(gridDim.x * blockDim.x);
    for (int i = (int)(blockIdx.x * blockDim.x + threadIdx.x); i < n4; i += stride) {
        out[i] = in[i];
    }
}

// ---- Kernel 2: one workgroup per patch; async-stage 64KB patch into LDS,
// then hardware-f32-atomic-add into the scattered output block. ----
__global__ __launch_bounds__(TPB) void scatter2dc_scatter_kernel(
    const float* __restrict__ gathered,
    const long long* __restrict__ indices,
    float* __restrict__ out) {
    __shared__ float tile[PATCH_F];            // 64 KB
    const int n = (int)blockIdx.x;
    const float* src = gathered + (size_t)n * PATCH_F;

    // Issue 16 async b128 transfers per thread: 256 threads x 16 x 16B = 64 KB.
    #pragma unroll
    for (int k = 0; k < PATCH_F4 / TPB; ++k) {
        const int e = ((int)threadIdx.x + k * TPB) * 4;  // float index, 16B aligned
        async_g2l_b128(src + e, &tile[e]);
    }

    // Overlap: compute destination base while the async engine streams LDS.
    const long long b  = indices[3 * n + 0];
    const long long hi = indices[3 * n + 1];
    const long long wi = indices[3 * n + 2];
    const size_t obase = (size_t)b * (CC * (size_t)HH * WW)
                       + (size_t)hi * (KH * (size_t)WW)
                       + (size_t)wi * KW;

    wait_async_done();      // this wave's async transfers landed in LDS
    __syncthreads();        // all waves' transfers visible

    // Each float4 in the patch: fid -> (channel c, row i, 4-col group j4).
    #pragma unroll
    for (int k = 0; k < PATCH_F4 / TPB; ++k) {
        const int fid = (int)threadIdx.x + k * TPB;   // [0, 4096)
        const int c   = fid >> 6;                     // 64 float4 per channel
        const int rem = fid & 63;
        const int i   = rem >> 2;                     // 4 float4 per row
        const int j4  = rem & 3;
        const float4 v = *(const float4*)&tile[fid * 4];
        float* dst = out + obase + (size_t)c * (HH * (size_t)WW)
                                 + (size_t)i * WW + j4 * 4;
        // Duplicated patch indices overlap -> must accumulate atomically.
        unsafeAtomicAdd(dst + 0, v.x);
        unsafeAtomicAdd(dst + 1, v.y);
        unsafeAtomicAdd(dst + 2, v.z);
        unsafeAtomicAdd(dst + 3, v.w);
    }
}

extern "C" void kernel_launch(void* const* d_in, const int* in_sizes, int n_in,
                              void* d_out, int out_size, void* d_ws, size_t ws_size,
                              hipStream_t stream) {
    const float*     inp  = (const float*)d_in[0];
    const float*     gath = (const float*)d_in[1];
    const long long* idx  = (const long long*)d_in[2];
    float*           out  = (float*)d_out;

    // Phase 1: out = input. 67.1M floats = 16.8M float4.
    const int n4 = out_size / 4;
    scatter2dc_copy_kernel<<<8192, TPB, 0, stream>>>(
        (const float4*)inp, (float4*)out, n4);

    // Phase 2: scatter-add N patches (stream order gives the copy->atomic dependency).
    const int N = in_sizes[2] / 3;   // indices is [N,3] int64
    scatter2dc_scatter_kernel<<<N, TPB, 0, stream>>>(gath, idx, out);
}